// WaveletScatteringTransform_25580825215398
// MI455X (gfx1250) — compile-verified
//
#include <hip/hip_runtime.h>
#include <math.h>

// ---------------------------------------------------------------------------
// Wavelet scattering on gfx1250: every 16000-pt (I)FFT is done as two complex
// GEMM stages (16000 = 128 x 125 four-step) with V_WMMA_F32_16X16X4_F32,
// entirely inside one workgroup's LDS (263 KB of the 320 KB WGP budget).
// Ragged 125-vs-128 edges are handled by zero-padded symmetric tables and a
// transposed, stride-132 Y buffer, so the WMMA loops contain no EXEC-masked
// accesses, no strided global pairs, and no LDS bank conflicts.
// ---------------------------------------------------------------------------

namespace {
constexpr int TLEN  = 16000;
constexpr int R1    = 128;       // radix-128 stage (F128 is 128x128)
constexpr int R2    = 125;       // radix-125 stage
constexpr int PS    = 128;       // padded stride for F125/TW and output rows
constexpr int YS    = 132;       // transposed-Y LDS stride (132 % 64 = 4)
constexpr int PROW  = 16384;     // padded global spectrum row stride
constexpr int LPRAD = 512;       // low-pass time support (+-5 sigma_t)
constexpr int LPLEN = 2 * LPRAD + 1;
constexpr int NF1   = 48;        // J*Q first-order wavelets
constexpr int NPATH = 120;       // second-order paths (j2 > j1)
constexpr int NCH   = 169;       // 1 + 48 + 120
constexpr int TSUB  = 250;       // T / 64
constexpr int SUB   = 64;
constexpr float PI2 = 6.28318530717958647692f;
}

typedef float v2f __attribute__((ext_vector_type(2)));
typedef float v8f __attribute__((ext_vector_type(8)));

// D = A(16x4 f32) * B(4x16 f32) + C, full fp32 — the precision an FFT needs.
__device__ __forceinline__ v8f wmma4(v2f a, v2f b, v8f c) {
  return __builtin_amdgcn_wmma_f32_16x16x4_f32(
      /*neg_a=*/false, a, /*neg_b=*/false, b,
      /*c_mod=*/(short)0, c, /*reuse_a=*/false, /*reuse_b=*/false);
}

// ---------------------------------------------------------------------------
// Stage A:  Y[k1][n2] = tw[k1][n2] * sum_n1 F128[k1][n1] * u[125*n1 + n2]
//   A = F128 (global, L2 resident, contiguous kk -> b64), B = u in LDS
//   (ds_load_2addr, offset pair = 125).  Y is written TRANSPOSED (YT[n2][k1],
//   stride 132) so the j-loop store is 2x ds_store_b128 without conflicts.
//   TW is symmetric -> read TW[nn][mm] contiguously in j (b128 merge).
//   Pad columns n in [125,128) read <=3 floats past u[16000) (finite by
//   construction); they land in YT pad rows, annihilated in stage B by the
//   zero-padded F125.  cj!=0 -> conjugated tables (inverse transform).
// ---------------------------------------------------------------------------
__device__ __forceinline__ void dft_stageA(
    const float* __restrict__ Fr, const float* __restrict__ Fi, int cj,
    const float* __restrict__ TWr, const float* __restrict__ TWi,
    const float* Br, const float* Bi, int hasImag,
    float* Yr, float* Yi)
{
  const int lane = threadIdx.x & 31;
  const int wv   = threadIdx.x >> 5;   // 8 waves -> 8 row-bands of 16
  const int loc  = lane & 15;
  const int hl   = lane >> 4;          // 0/1: which K-pair / which C half
  const int m    = wv * 16 + loc;      // A-operand row for this lane
  const int mb   = wv * 16 + hl * 8;   // C-tile row base for this lane

  for (int nt = 0; nt < 8; ++nt) {
    const int n = nt * 16 + loc;       // B-operand column for this lane
    v8f cr = {0.f, 0.f, 0.f, 0.f, 0.f, 0.f, 0.f, 0.f};
    v8f ci = {0.f, 0.f, 0.f, 0.f, 0.f, 0.f, 0.f, 0.f};

    for (int k0 = 0; k0 < R1; k0 += 4) {
      const int kk = k0 + hl * 2;
      v2f ar, ai, nai, br, bi;
      ar.x = Fr[m * R1 + kk];
      ar.y = Fr[m * R1 + kk + 1];
      float aix = Fi[m * R1 + kk];
      float aiy = Fi[m * R1 + kk + 1];
      if (cj) { aix = -aix; aiy = -aiy; }   // f32 WMMA has no A-neg: do in VALU
      ai.x = aix;  ai.y = aiy;
      nai.x = -aix; nai.y = -aiy;
      br.x = Br[kk * R2 + n];
      br.y = Br[(kk + 1) * R2 + n];
      cr = wmma4(ar, br, cr);               // Re += Ar*Br
      ci = wmma4(ai, br, ci);               // Im += Ai*Br
      if (hasImag) {
        bi.x = Bi[kk * R2 + n];
        bi.y = Bi[(kk + 1) * R2 + n];
        cr = wmma4(nai, bi, cr);            // Re -= Ai*Bi
        ci = wmma4(ar, bi, ci);             // Im += Ar*Bi
      }
    }
    const int nn = nt * 16 + loc;
    const float* twrp = TWr + nn * PS + mb;   // symmetric table: TW[nn][mm]
    const float* twip = TWi + nn * PS + mb;
    float* yrp = Yr + nn * YS + mb;           // transposed Y: YT[nn][mm]
    float* yip = Yi + nn * YS + mb;
#pragma unroll
    for (int j = 0; j < 8; ++j) {             // C VGPR j <-> row mb + j
      float twr = twrp[j];
      float twi = twip[j];
      if (cj) twi = -twi;
      const float vr = cr[j], vi = ci[j];
      yrp[j] = vr * twr - vi * twi;
      yip[j] = vr * twi + vi * twr;
    }
  }
}

// ---------------------------------------------------------------------------
// Stage B:  X[k1 + 128*k2] = sum_n2 Y[k1][n2] * F125[n2][k2]
//   A = YT in LDS (A[m][k] = YT[k*132+m], ds_load_2addr pair offset 132,
//   conflict-free), B = zero-padded symmetric F125 read as F125[n][kk]
//   (contiguous kk -> b64; rows/cols >=125 are 0, killing YT's pad rows).
//   Output rows padded to 16384 -> unconditional contiguous stores.
// ---------------------------------------------------------------------------
__device__ __forceinline__ void dft_stageB(
    const float* Yr, const float* Yi,
    const float* __restrict__ Fr, const float* __restrict__ Fi, int cj,
    float scale, int doAbs, float* outR, float* outI)
{
  const int lane = threadIdx.x & 31;
  const int wv   = threadIdx.x >> 5;
  const int loc  = lane & 15;
  const int hl   = lane >> 4;
  const int m    = wv * 16 + loc;
  const int mb   = wv * 16 + hl * 8;

  for (int nt = 0; nt < 8; ++nt) {
    const int n = nt * 16 + loc;
    v8f cr = {0.f, 0.f, 0.f, 0.f, 0.f, 0.f, 0.f, 0.f};
    v8f ci = {0.f, 0.f, 0.f, 0.f, 0.f, 0.f, 0.f, 0.f};

    for (int k0 = 0; k0 < R1; k0 += 4) {    // K padded to 128 (zeros in F125)
      const int kk = k0 + hl * 2;
      v2f ar, ai, nai, br, bi;
      ar.x = Yr[kk * YS + m];               // YT[k][m]
      ar.y = Yr[(kk + 1) * YS + m];
      const float aix = Yi[kk * YS + m];
      const float aiy = Yi[(kk + 1) * YS + m];
      ai.x = aix;  ai.y = aiy;
      nai.x = -aix; nai.y = -aiy;
      float fi0 = Fi[n * PS + kk];          // symmetric: F125[n][kk]
      float fi1 = Fi[n * PS + kk + 1];
      if (cj) { fi0 = -fi0; fi1 = -fi1; }
      br.x = Fr[n * PS + kk];
      br.y = Fr[n * PS + kk + 1];
      bi.x = fi0; bi.y = fi1;
      cr = wmma4(ar, br, cr);
      cr = wmma4(nai, bi, cr);
      ci = wmma4(ar, bi, ci);
      ci = wmma4(ai, br, ci);
    }
    const int nn = nt * 16 + loc;
    float* orp = outR + PS * nn + mb;       // linear k1 + 128*k2, contiguous j
    float* oip = doAbs ? nullptr : (outI + PS * nn + mb);
#pragma unroll
    for (int j = 0; j < 8; ++j) {
      const float vr = cr[j] * scale;
      const float vi = ci[j] * scale;
      if (doAbs) {
        orp[j] = sqrtf(vr * vr + vi * vi);
      } else {
        orp[j] = vr;
        oip[j] = vi;
      }
    }
  }
}

// Gaussian low-pass + subsample by 64.  Each lane starts its tap sweep at a
// lane-dependent rotation (lane*33) so per-lane LDS addresses stride 97 mod 64
// across the wave -> no 32-way bank conflicts on u[].
__device__ __forceinline__ void lowpass_store(
    const float* u, const float* __restrict__ taps, float* __restrict__ outp)
{
  const int d0 = (threadIdx.x & 31) * 33;   // < LPLEN
  for (int tau = threadIdx.x; tau < TSUB; tau += blockDim.x) {
    float acc = 0.f;
    int t = tau * SUB - LPRAD + d0;
    t %= TLEN; if (t < 0) t += TLEN;
    int d = d0;
    for (int it = 0; it < LPLEN; ++it) {
      acc += u[t] * taps[d];
      if (++t == TLEN) t = 0;
      if (++d == LPLEN) { d = 0; t -= LPLEN; if (t < 0) t += TLEN; }
    }
    outp[tau] = acc;
  }
}

// ---------------------------------------------------------------------------
// Table builder: F128 (128x128), zero-padded F125 (128x128), padded twiddles
// W_N^{k1*n2} (128x128; all symmetric since they depend on the index product),
// and the 1025 low-pass taps phi_t[d] = (1/T) sum_k phi_f[k] e^{2pi i k d / T}
// (phi_f sigma ~25 bins, so k<1024 with Hermitian doubling is exact to ~1e-8).
// ---------------------------------------------------------------------------
__global__ __launch_bounds__(256, 1) void k_tables(
    float* F128r, float* F128i, float* F125r, float* F125i,
    float* TWr, float* TWi, float* LP, const float* __restrict__ phi)
{
  int i = blockIdx.x * blockDim.x + threadIdx.x;
  if (i < R1 * R1) {
    const int k = i / R1, n = i % R1;
    const int m = (k * n) & (R1 - 1);
    const float th = PI2 * (float)m / (float)R1;
    F128r[i] = cosf(th); F128i[i] = -sinf(th);
  } else if ((i -= R1 * R1) < PS * PS) {
    const int k = i / PS, n = i % PS;
    if (k < R2 && n < R2) {
      const int m = (k * n) % R2;
      const float th = PI2 * (float)m / (float)R2;
      F125r[i] = cosf(th); F125i[i] = -sinf(th);
    } else {
      F125r[i] = 0.f; F125i[i] = 0.f;       // zero pad kills K in [125,128)
    }
  } else if ((i -= PS * PS) < PS * PS) {
    const int k1 = i / PS, n2 = i % PS;
    const int m = (k1 * n2) % TLEN;
    const float th = PI2 * (float)m / (float)TLEN;
    TWr[i] = cosf(th); TWi[i] = -sinf(th);
  } else if ((i -= PS * PS) < LPLEN) {
    const int delta = i - LPRAD;
    float acc = phi[0];
    for (int k = 1; k < 1024; ++k) {
      int m = (k * delta) % TLEN; if (m < 0) m += TLEN;
      acc += 2.0f * phi[k] * cosf(PI2 * (float)m / (float)TLEN);
    }
    LP[i] = acc / (float)TLEN;
  }
}

// ---------------------------------------------------------------------------
// Order 0 + forward FFT of windowed input:  xw -> S0, Xf
// ---------------------------------------------------------------------------
__global__ __launch_bounds__(256, 1) void k_fwd0(
    const float* __restrict__ x, const float* __restrict__ win,
    const float* __restrict__ F128r, const float* __restrict__ F128i,
    const float* __restrict__ F125r, const float* __restrict__ F125i,
    const float* __restrict__ TWr, const float* __restrict__ TWi,
    const float* __restrict__ LP,
    float* __restrict__ Xfr, float* __restrict__ Xfi, float* __restrict__ out)
{
  extern __shared__ float smem[];
  float* Br = smem;                        // 16000
  float* Bi = smem + TLEN;                 // 16000 (first 4 zeroed: overrun)
  float* Yr = smem + 2 * TLEN;             // 128*132 transposed
  float* Yi = smem + 2 * TLEN + R1 * YS;

  const int b = blockIdx.x;
  const float4* x4 = (const float4*)(x + (size_t)b * TLEN);
  const float4* w4 = (const float4*)win;
  float4* B4r = (float4*)Br;
  for (int i = threadIdx.x; i < TLEN / 4; i += blockDim.x) {
    if ((i & 511) == 0) __builtin_prefetch(x4 + i + 1024, 0, 0);
    const float4 a = x4[i], w = w4[i];
    float4 v;
    v.x = a.x * w.x; v.y = a.y * w.y; v.z = a.z * w.z; v.w = a.w * w.w;
    B4r[i] = v;
  }
  if (threadIdx.x < 4) Bi[threadIdx.x] = 0.f;   // stage-A tail overrun reads
  __syncthreads();

  lowpass_store(Br, LP, out + ((size_t)b * NCH + 0) * TSUB);          // S0
  dft_stageA(F128r, F128i, /*cj=*/0, TWr, TWi, Br, Bi, /*hasImag=*/0, Yr, Yi);
  __syncthreads();
  dft_stageB(Yr, Yi, F125r, F125i, /*cj=*/0, 1.0f, /*doAbs=*/0,
             Xfr + (size_t)b * PROW, Xfi + (size_t)b * PROW);         // Xf
}

// ---------------------------------------------------------------------------
// Order 1, fully fused per (b, n):
//   U1 = |IFFT(Xf * psi1[n])| (stays in LDS) -> S1[b,1+n,:], U1f[b,n,:]
// ---------------------------------------------------------------------------
__global__ __launch_bounds__(256, 1) void k_o1(
    const float* __restrict__ Xfr, const float* __restrict__ Xfi,
    const float* __restrict__ psi1,
    const float* __restrict__ F128r, const float* __restrict__ F128i,
    const float* __restrict__ F125r, const float* __restrict__ F125i,
    const float* __restrict__ TWr, const float* __restrict__ TWi,
    const float* __restrict__ LP,
    float* __restrict__ U1fr, float* __restrict__ U1fi, float* __restrict__ out)
{
  extern __shared__ float smem[];
  float* Br = smem;
  float* Bi = smem + TLEN;
  float* Yr = smem + 2 * TLEN;
  float* Yi = smem + 2 * TLEN + R1 * YS;

  const int z = blockIdx.x;
  const int b = z / NF1;
  const int n = z % NF1;
  const float4* s4r = (const float4*)(Xfr + (size_t)b * PROW);
  const float4* s4i = (const float4*)(Xfi + (size_t)b * PROW);
  const float4* f4  = (const float4*)(psi1 + (size_t)n * TLEN);
  float4* B4r = (float4*)Br;
  float4* B4i = (float4*)Bi;
  for (int i = threadIdx.x; i < TLEN / 4; i += blockDim.x) {
    if ((i & 511) == 0) __builtin_prefetch(f4 + i + 1024, 0, 0);
    const float4 fv = f4[i], a = s4r[i], c = s4i[i];
    float4 vr, vi;
    vr.x = a.x * fv.x; vr.y = a.y * fv.y; vr.z = a.z * fv.z; vr.w = a.w * fv.w;
    vi.x = c.x * fv.x; vi.y = c.y * fv.y; vi.z = c.z * fv.z; vi.w = c.w * fv.w;
    B4r[i] = vr;
    B4i[i] = vi;
  }
  __syncthreads();

  // inverse transform (conjugated tables), |.|/N into Br (pads land in Bi[:384])
  dft_stageA(F128r, F128i, 1, TWr, TWi, Br, Bi, 1, Yr, Yi);
  __syncthreads();
  dft_stageB(Yr, Yi, F125r, F125i, 1, 1.0f / (float)TLEN, /*doAbs=*/1, Br, Bi);
  __syncthreads();

  lowpass_store(Br, LP, out + ((size_t)b * NCH + 1 + n) * TSUB);      // S1

  // forward FFT of the (real) envelope -> U1f
  dft_stageA(F128r, F128i, 0, TWr, TWi, Br, Bi, /*hasImag=*/0, Yr, Yi);
  __syncthreads();
  dft_stageB(Yr, Yi, F125r, F125i, 0, 1.0f, /*doAbs=*/0,
             U1fr + ((size_t)b * NF1 + n) * PROW,
             U1fi + ((size_t)b * NF1 + n) * PROW);
}

// ---------------------------------------------------------------------------
// Order 2, fully fused per (b, p):  S2[b,49+p,:] = LP|IFFT(U1f[i1]*psi2[i2])|
// ---------------------------------------------------------------------------
__global__ __launch_bounds__(256, 1) void k_o2(
    const float* __restrict__ U1fr, const float* __restrict__ U1fi,
    const float* __restrict__ psi2,
    const int* __restrict__ i1_idx, const int* __restrict__ i2_idx,
    const float* __restrict__ F128r, const float* __restrict__ F128i,
    const float* __restrict__ F125r, const float* __restrict__ F125i,
    const float* __restrict__ TWr, const float* __restrict__ TWi,
    const float* __restrict__ LP, float* __restrict__ out)
{
  extern __shared__ float smem[];
  float* Br = smem;
  float* Bi = smem + TLEN;
  float* Yr = smem + 2 * TLEN;
  float* Yi = smem + 2 * TLEN + R1 * YS;

  const int z = blockIdx.x;
  const int b = z / NPATH;
  const int p = z % NPATH;
  const int n1 = i1_idx[p];
  const int j2 = i2_idx[p];
  const float4* s4r = (const float4*)(U1fr + ((size_t)b * NF1 + n1) * PROW);
  const float4* s4i = (const float4*)(U1fi + ((size_t)b * NF1 + n1) * PROW);
  const float4* f4  = (const float4*)(psi2 + (size_t)j2 * TLEN);
  float4* B4r = (float4*)Br;
  float4* B4i = (float4*)Bi;
  for (int i = threadIdx.x; i < TLEN / 4; i += blockDim.x) {
    if ((i & 511) == 0) __builtin_prefetch(s4r + i + 1024, 0, 0);
    const float4 fv = f4[i], a = s4r[i], c = s4i[i];
    float4 vr, vi;
    vr.x = a.x * fv.x; vr.y = a.y * fv.y; vr.z = a.z * fv.z; vr.w = a.w * fv.w;
    vi.x = c.x * fv.x; vi.y = c.y * fv.y; vi.z = c.z * fv.z; vi.w = c.w * fv.w;
    B4r[i] = vr;
    B4i[i] = vi;
  }
  __syncthreads();

  dft_stageA(F128r, F128i, 1, TWr, TWi, Br, Bi, 1, Yr, Yi);
  __syncthreads();
  dft_stageB(Yr, Yi, F125r, F125i, 1, 1.0f / (float)TLEN, /*doAbs=*/1, Br, Bi);
  __syncthreads();

  lowpass_store(Br, LP, out + ((size_t)b * NCH + 1 + NF1 + p) * TSUB);
}

// ---------------------------------------------------------------------------
extern "C" void kernel_launch(void* const* d_in, const int* in_sizes, int n_in,
                              void* d_out, int out_size, void* d_ws, size_t ws_size,
                              hipStream_t stream) {
  const float* x    = (const float*)d_in[0];
  const float* win  = (const float*)d_in[1];
  const float* psi1 = (const float*)d_in[2];
  const float* psi2 = (const float*)d_in[3];
  const float* phi  = (const float*)d_in[4];
  const int*   i1   = (const int*)d_in[5];
  const int*   i2   = (const int*)d_in[6];
  float* out = (float*)d_out;
  float* ws  = (float*)d_ws;

  const int B = in_sizes[0] / TLEN;   // 32

  // workspace layout (f32): tables + Xf + U1f (everything else lives in LDS)
  float* F128r = ws;                  float* F128i = F128r + R1 * R1;
  float* F125r = F128i + R1 * R1;     float* F125i = F125r + PS * PS;
  float* TWr   = F125i + PS * PS;     float* TWi   = TWr + PS * PS;
  float* LP    = TWi + PS * PS;
  float* Xfr   = LP + LPLEN;          float* Xfi   = Xfr + (size_t)B * PROW;
  float* U1fr  = Xfi + (size_t)B * PROW;
  float* U1fi  = U1fr + (size_t)B * NF1 * PROW;

  // LDS: Br(16000) Bi(16000) Yr(128*132) Yi(128*132) = 263 KB of 320 KB WGP
  const size_t lds = (2 * (size_t)TLEN + 2 * (size_t)R1 * YS) * sizeof(float);
  const int tblN = R1 * R1 + PS * PS + PS * PS + LPLEN;

  k_tables<<<(tblN + 255) / 256, 256, 0, stream>>>(
      F128r, F128i, F125r, F125i, TWr, TWi, LP, phi);
  k_fwd0<<<B, 256, lds, stream>>>(
      x, win, F128r, F128i, F125r, F125i, TWr, TWi, LP, Xfr, Xfi, out);
  k_o1<<<B * NF1, 256, lds, stream>>>(
      Xfr, Xfi, psi1, F128r, F128i, F125r, F125i, TWr, TWi, LP,
      U1fr, U1fi, out);
  k_o2<<<B * NPATH, 256, lds, stream>>>(
      U1fr, U1fi, psi2, i1, i2, F128r, F128i, F125r, F125i, TWr, TWi, LP, out);
}